// tensor_display_11897059410368
// MI455X (gfx1250) — compile-verified
//
#include <hip/hip_runtime.h>

typedef __attribute__((ext_vector_type(2))) float v2f;
typedef __attribute__((ext_vector_type(8))) float v8f;

#define HW (512 * 512)

// One wave computes one (n, hb, wb) 64x64 outer-product block:
//   A = back block  (64 x 4, K = 3 channels + 1 zero pad), pre-scaled by 1/3
//   B = front block (4 x 64)
//   D = A x B, scattered to out.
__global__ __launch_bounds__(256) void outer_block_wmma_kernel(
    const float* __restrict__ layers, float* __restrict__ out)
{
    const int lane = threadIdx.x & 31;
    const int wave = threadIdx.x >> 5;
    const int half = lane >> 4;   // 0 -> K holds channels {0,1}; 1 -> {2, pad}
    const int lm   = lane & 15;

    // One wave per block: blk in [0, 8*64*64)
    const int blk = blockIdx.x * 8 + wave;
    const int n   = blk >> 12;        // / 4096
    const int hb  = (blk >> 6) & 63;
    const int wb  = blk & 63;

    const float* back  = layers + (size_t)((n * 2 + 0) * 3) * HW;
    const float* front = layers + (size_t)((n * 2 + 1) * 3) * HW;

    const int c0 = half ? 2 : 0;      // .x channel
    const int c1 = half ? 0 : 1;      // .y channel source (always in-bounds)
    const float inv_c = 1.0f / 3.0f;
    const float sy = half ? 0.0f : inv_c;   // zero the K=3 pad via the scale
    const float by = half ? 0.0f : 1.0f;

    // A fragments (back, pre-scaled by 1/3): 4 M-tiles of 16 rows (m = i*8+j)
    v2f a[4];
#pragma unroll
    for (int t = 0; t < 4; ++t) {
        const int m = t * 16 + lm;
        const size_t off = (size_t)(hb * 8 + (m >> 3)) * 512 + (wb * 8 + (m & 7));
        const float xv = back[(size_t)c0 * HW + off];
        const float yv = back[(size_t)c1 * HW + off];   // unconditional, no branch
        a[t].x = xv * inv_c;
        a[t].y = yv * sy;                               // cndmask-zeroed pad
    }

    // B fragments (front): 4 N-tiles of 16 cols (p = k*8+l)
    v2f b[4];
#pragma unroll
    for (int s = 0; s < 4; ++s) {
        const int p = s * 16 + lm;
        const size_t off = (size_t)(hb * 8 + (p >> 3)) * 512 + (wb * 8 + (p & 7));
        const float xv = front[(size_t)c0 * HW + off];
        const float yv = front[(size_t)c1 * HW + off];  // unconditional, no branch
        b[s].x = xv;
        b[s].y = yv * by;                               // cndmask-zeroed pad
    }

#pragma unroll
    for (int t = 0; t < 4; ++t) {
#pragma unroll
        for (int s = 0; s < 4; ++s) {
            v8f d = {};
            // D = A_t x B_s  (16x16x4 f32 WMMA; K=3 real + zero pad)
            d = __builtin_amdgcn_wmma_f32_16x16x4_f32(
                /*neg_a=*/false, a[t], /*neg_b=*/false, b[s],
                /*c_mod=*/(short)0, d, /*reuse_a=*/false, /*reuse_b=*/false);

            const int p   = s * 16 + lm;
            const int row = hb * 8 + (p >> 3);
            const int col = wb * 8 + (p & 7);
            float* base = out + (((size_t)n * 64 + t * 16 + half * 8) * 512 + row) * 512 + col;
#pragma unroll
            for (int r = 0; r < 8; ++r) {
                // consecutive r -> consecutive output channel planes (HW apart)
                base[(size_t)r * HW] = d[r];
            }
        }
    }
}

extern "C" void kernel_launch(void* const* d_in, const int* in_sizes, int n_in,
                              void* d_out, int out_size, void* d_ws, size_t ws_size,
                              hipStream_t stream)
{
    const float* layers = (const float*)d_in[0];
    float* out = (float*)d_out;

    // 8 * 64 * 64 = 32768 blocks, 8 waves (256 threads) per workgroup
    const int num_blocks = 8 * 64 * 64;
    const dim3 grid(num_blocks / 8);
    const dim3 block(256);
    outer_block_wmma_kernel<<<grid, block, 0, stream>>>(layers, out);
}